// DifferentiableTVLayer_39522289058098
// MI455X (gfx1250) — compile-verified
//
#include <hip/hip_runtime.h>
#include <stdint.h>

// ---------------------------------------------------------------------------
// Chambolle-Pock anisotropic TV prox, B=8, H=W=256, 200 iterations.
//
// MI455X / gfx1250 design:
//  * 1 workgroup (1024 threads = 32 wave32 waves) per batch image -> all 400
//    per-iteration syncs are in-WGP split barriers.
//  * ubar (256KB + edge pad) lives in LDS: the only array with
//    latency-critical cross-thread reads every iteration.
//  * No persistent per-thread register arrays (VGPR budget at 1024 threads
//    cannot hold them): p,q live in d_ws, u in d_out, all L2-resident
//    (1.25MB/image incl. f,lam vs 192MB L2). Cross-thread dual halos are plain global
//    reads. Peak live VGPRs ~= 70 row transients -> no spills.
//  * Row loops are NOT unrolled: the 200x hot loop stays small (I-cache /
//    prefetch friendly); latency hiding comes from 8 waves/SIMD, and all
//    row accesses remain b128 vector ops.
//  * ubar0 = f bulk-loaded with the Tensor Data Mover (tensor_load_to_lds +
//    s_wait_tensorcnt), D# assembled per ISA 8.3/8.4.
//  * Invalid duals (p row 255 / q col 255) are written as +-0 by clipping
//    against lam = 0, keeping the inner loops branch-free and uniform.
// ---------------------------------------------------------------------------

#define HDIM 256
#define WDIM 256
#define NITER 200
#define TAUC 0.35355339f
#define SIGC 0.35355339f
#define IMG_ELEMS (HDIM * WDIM)

// LDS: ubar image + 272 floats of pad so the deliberate last-row/last-col
// over-reads (whose results are clipped away) stay inside the allocation.
#define LDS_FLOATS (IMG_ELEMS + 272)
#define LDS_BYTES  (LDS_FLOATS * sizeof(float))

typedef unsigned int uv4 __attribute__((ext_vector_type(4)));
typedef int          iv4 __attribute__((ext_vector_type(4)));
typedef int          iv8 __attribute__((ext_vector_type(8)));

__global__ __launch_bounds__(1024, 1)
void tv_chambolle_pock(const float* __restrict__ f_g,
                       const float* __restrict__ lam_g,
                       float* __restrict__ u_g,     // d_out: u, final output
                       float* __restrict__ p_ws,    // d_ws: p duals (8 images)
                       float* __restrict__ q_ws)    // d_ws: q duals (8 images)
{
  extern __shared__ float s_ubar[];   // [256][256] + pad

  const int b   = blockIdx.x;
  const int tid = threadIdx.x;
  const int tx  = tid & 31;           // tile column 0..31 (lane id)
  const int ty  = tid >> 5;           // tile row    0..31 (wave-uniform)
  const int gj0 = tx << 3;
  const int gi0 = ty << 3;

  const float* __restrict__ fb   = f_g   + (size_t)b * IMG_ELEMS;
  const float* __restrict__ lamb = lam_g + (size_t)b * IMG_ELEMS;
  float* __restrict__ ub_g = u_g  + (size_t)b * IMG_ELEMS;
  float* __restrict__ pb   = p_ws + (size_t)b * IMG_ELEMS;
  float* __restrict__ qb   = q_ws + (size_t)b * IMG_ELEMS;

  // ---- init: ubar0 = f via Tensor Data Mover DMA --------------------------
#if defined(__has_builtin) && __has_builtin(__builtin_amdgcn_tensor_load_to_lds) && __has_builtin(__builtin_amdgcn_s_wait_tensorcnt)
  if (tid < 32) {   // one wave issues the TDM op (descriptor is wave-scalar)
    const uint64_t ga  = (uint64_t)(uintptr_t)fb;
    const uint32_t lds = (uint32_t)(uintptr_t)s_ubar;  // low 32 bits = LDS offset

    uv4 g0;
    g0[0] = 1u;                                  // count=1, user mode, no gather
    g0[1] = lds;                                 // lds_addr
    g0[2] = (uint32_t)ga;                        // global_addr[31:0]
    g0[3] = (uint32_t)((ga >> 32) & 0x01FFFFFFu) // global_addr[56:32]
            | (2u << 30);                        // type = 2 ("image")

    iv8 g1;
    g1[0] = 0x00020000;          // workgroup_mask=0, data_size=2 (4 bytes)
    g1[1] = (int)(WDIM << 16);   // tensor_dim0 = 256 (bits[79:48], low half)
    g1[2] = (int)(HDIM << 16);   // tensor_dim1 = 256 (bits[111:80], low half)
    g1[3] = (int)(WDIM << 16);   // tile_dim0 = 256 (bits[127:112])
    g1[4] = (int)HDIM;           // tile_dim1 = 256, tile_dim2 = 0
    g1[5] = (int)WDIM;           // tensor_dim0_stride = 256 (low 32)
    g1[6] = 0;
    g1[7] = 0;

    iv4 g2 = {0, 0, 0, 0};       // 2D tensor: groups 2/3 unused
    iv4 g3 = {0, 0, 0, 0};
    iv8 g4 = {0, 0, 0, 0, 0, 0, 0, 0};

    __builtin_amdgcn_tensor_load_to_lds(g0, g1, g2, g3, g4, 0);
    __builtin_amdgcn_s_wait_tensorcnt(0);
  }
#else
  for (int k = tid; k < IMG_ELEMS; k += 1024) s_ubar[k] = fb[k];
#endif
  __syncthreads();

  // ---- u0 = f (LDS -> d_out), p0 = q0 = 0 (d_ws may hold poison) ----------
  const float4 z4 = make_float4(0.f, 0.f, 0.f, 0.f);
#pragma unroll 1
  for (int ii = 0; ii < 8; ++ii) {
    const int r = (gi0 + ii) * WDIM + gj0;
    const float4 a = *reinterpret_cast<const float4*>(&s_ubar[r]);
    const float4 c = *reinterpret_cast<const float4*>(&s_ubar[r + 4]);
    *reinterpret_cast<float4*>(&ub_g[r])     = a;
    *reinterpret_cast<float4*>(&ub_g[r + 4]) = c;
    *reinterpret_cast<float4*>(&pb[r])     = z4;
    *reinterpret_cast<float4*>(&pb[r + 4]) = z4;
    *reinterpret_cast<float4*>(&qb[r])     = z4;
    *reinterpret_cast<float4*>(&qb[r + 4]) = z4;
  }

  const float inv1pt = 1.0f / (1.0f + TAUC);

#pragma clang loop unroll(disable)
  for (int it = 0; it < NITER; ++it) {
    // =====================  Phase A: dual update  ==========================
    // p[i,j]=clip(p+sig*(ubar[i+1,j]-ubar[i,j]), +-lam[i+1,j])
    // q[i,j]=clip(q+sig*(ubar[i,j+1]-ubar[i,j]), +-lam[i,j+1])
    float rc[9], lc[9];
    {
      const float* row = &s_ubar[gi0 * WDIM + gj0];
      const float4 a = *reinterpret_cast<const float4*>(row);
      const float4 c = *reinterpret_cast<const float4*>(row + 4);
      rc[0] = a.x; rc[1] = a.y; rc[2] = a.z; rc[3] = a.w;
      rc[4] = c.x; rc[5] = c.y; rc[6] = c.z; rc[7] = c.w;
      rc[8] = row[8];   // col gj0+8 (row gi0+1 col 0 for tx==31; guarded below)
    }
    {
      const float4 a = *reinterpret_cast<const float4*>(&lamb[gi0 * WDIM + gj0]);
      const float4 c = *reinterpret_cast<const float4*>(&lamb[gi0 * WDIM + gj0 + 4]);
      lc[0] = a.x; lc[1] = a.y; lc[2] = a.z; lc[3] = a.w;
      lc[4] = c.x; lc[5] = c.y; lc[6] = c.z; lc[7] = c.w;
      lc[8] = lamb[gi0 * WDIM + gj0 + 8];  // gi0 <= 248 -> inside image b
    }
#pragma unroll 1
    for (int ii = 0; ii < 8; ++ii) {
      const int gi = gi0 + ii;
      const int r  = gi * WDIM + gj0;
      float rn[9];
      {
        const float* row = &s_ubar[r + WDIM];   // row gi+1 (pad at gi==255)
        const float4 a = *reinterpret_cast<const float4*>(row);
        const float4 c = *reinterpret_cast<const float4*>(row + 4);
        rn[0] = a.x; rn[1] = a.y; rn[2] = a.z; rn[3] = a.w;
        rn[4] = c.x; rn[5] = c.y; rn[6] = c.z; rn[7] = c.w;
        rn[8] = row[8];
      }
      float ln[9];
      if (gi < HDIM - 1) {   // wave-uniform branch
        const float4 a = *reinterpret_cast<const float4*>(&lamb[r + WDIM]);
        const float4 c = *reinterpret_cast<const float4*>(&lamb[r + WDIM + 4]);
        ln[0] = a.x; ln[1] = a.y; ln[2] = a.z; ln[3] = a.w;
        ln[4] = c.x; ln[5] = c.y; ln[6] = c.z; ln[7] = c.w;
        const bool safe = (gj0 + 8 < WDIM) || (gi + 1 < HDIM - 1);
        ln[8] = safe ? lamb[r + WDIM + 8] : 0.0f;
      } else {
#pragma unroll
        for (int k = 0; k < 9; ++k) ln[k] = 0.0f;
      }
      // ---- p row RMW (global, own row) ----
      {
        float pr[8];
        const float4 a = *reinterpret_cast<const float4*>(&pb[r]);
        const float4 c = *reinterpret_cast<const float4*>(&pb[r + 4]);
        pr[0] = a.x; pr[1] = a.y; pr[2] = a.z; pr[3] = a.w;
        pr[4] = c.x; pr[5] = c.y; pr[6] = c.z; pr[7] = c.w;
#pragma unroll
        for (int jj = 0; jj < 8; ++jj) {
          const float plam = ln[jj];                    // lam[gi+1][gj]
          float pv = pr[jj] + SIGC * (rn[jj] - rc[jj]);
          pr[jj] = fminf(fmaxf(pv, -plam), plam);
        }
        *reinterpret_cast<float4*>(&pb[r])     = make_float4(pr[0], pr[1], pr[2], pr[3]);
        *reinterpret_cast<float4*>(&pb[r + 4]) = make_float4(pr[4], pr[5], pr[6], pr[7]);
      }
      // ---- q row RMW (global, own row) ----
      {
        float qr[8];
        const float4 a = *reinterpret_cast<const float4*>(&qb[r]);
        const float4 c = *reinterpret_cast<const float4*>(&qb[r + 4]);
        qr[0] = a.x; qr[1] = a.y; qr[2] = a.z; qr[3] = a.w;
        qr[4] = c.x; qr[5] = c.y; qr[6] = c.z; qr[7] = c.w;
#pragma unroll
        for (int jj = 0; jj < 8; ++jj) {
          const float qlam = (gj0 + jj < WDIM - 1) ? lc[jj + 1] : 0.0f; // lam[gi][gj+1]
          float qv = qr[jj] + SIGC * (rc[jj + 1] - rc[jj]);
          qr[jj] = fminf(fmaxf(qv, -qlam), qlam);
        }
        *reinterpret_cast<float4*>(&qb[r])     = make_float4(qr[0], qr[1], qr[2], qr[3]);
        *reinterpret_cast<float4*>(&qb[r + 4]) = make_float4(qr[4], qr[5], qr[6], qr[7]);
      }
#pragma unroll
      for (int k = 0; k < 9; ++k) { rc[k] = rn[k]; lc[k] = ln[k]; }
    }

    __threadfence_block();   // make p/q stores WG-visible
    __syncthreads();         // duals ready; all ubar reads complete

    // =====================  Phase B: primal update  ========================
    // u_new=(u - tau*((p[i-1]-p[i]) + (q[j-1]-q[j])) + tau*f)/(1+tau)
    // ubar=2*u_new-u
    float ppr[8];            // p row gi-1 (zeros at the top image border)
    if (ty > 0) {            // wave-uniform branch
      const int r = (gi0 - 1) * WDIM + gj0;
      const float4 a = *reinterpret_cast<const float4*>(&pb[r]);
      const float4 c = *reinterpret_cast<const float4*>(&pb[r + 4]);
      ppr[0] = a.x; ppr[1] = a.y; ppr[2] = a.z; ppr[3] = a.w;
      ppr[4] = c.x; ppr[5] = c.y; ppr[6] = c.z; ppr[7] = c.w;
    } else {
#pragma unroll
      for (int k = 0; k < 8; ++k) ppr[k] = 0.0f;
    }
#pragma unroll 1
    for (int ii = 0; ii < 8; ++ii) {
      const int gi = gi0 + ii;
      const int r  = gi * WDIM + gj0;
      float pcr[8], qr[8], fr[8], uo[8];
      {
        const float4 a = *reinterpret_cast<const float4*>(&pb[r]);
        const float4 c = *reinterpret_cast<const float4*>(&pb[r + 4]);
        pcr[0] = a.x; pcr[1] = a.y; pcr[2] = a.z; pcr[3] = a.w;
        pcr[4] = c.x; pcr[5] = c.y; pcr[6] = c.z; pcr[7] = c.w;
      }
      {
        const float4 a = *reinterpret_cast<const float4*>(&qb[r]);
        const float4 c = *reinterpret_cast<const float4*>(&qb[r + 4]);
        qr[0] = a.x; qr[1] = a.y; qr[2] = a.z; qr[3] = a.w;
        qr[4] = c.x; qr[5] = c.y; qr[6] = c.z; qr[7] = c.w;
      }
      // q left halo: q[gi][gj0-1] from the left neighbor (0 at image border)
      float qlft0 = qb[r + ((tx > 0) ? -1 : 0)];
      qlft0 = (tx > 0) ? qlft0 : 0.0f;
      {
        const float4 a = *reinterpret_cast<const float4*>(&fb[r]);
        const float4 c = *reinterpret_cast<const float4*>(&fb[r + 4]);
        fr[0] = a.x; fr[1] = a.y; fr[2] = a.z; fr[3] = a.w;
        fr[4] = c.x; fr[5] = c.y; fr[6] = c.z; fr[7] = c.w;
      }
      {
        const float4 a = *reinterpret_cast<const float4*>(&ub_g[r]);
        const float4 c = *reinterpret_cast<const float4*>(&ub_g[r + 4]);
        uo[0] = a.x; uo[1] = a.y; uo[2] = a.z; uo[3] = a.w;
        uo[4] = c.x; uo[5] = c.y; uo[6] = c.z; uo[7] = c.w;
      }
      float un[8], ubv[8];
#pragma unroll
      for (int jj = 0; jj < 8; ++jj) {
        const float qlft = (jj > 0) ? qr[jj - 1] : qlft0;
        const float divg = (ppr[jj] - pcr[jj]) + (qlft - qr[jj]);
        un[jj]  = (uo[jj] - TAUC * divg + TAUC * fr[jj]) * inv1pt;
        ubv[jj] = 2.0f * un[jj] - uo[jj];
      }
      *reinterpret_cast<float4*>(&ub_g[r]) =
          make_float4(un[0], un[1], un[2], un[3]);
      *reinterpret_cast<float4*>(&ub_g[r + 4]) =
          make_float4(un[4], un[5], un[6], un[7]);
      *reinterpret_cast<float4*>(&s_ubar[r]) =
          make_float4(ubv[0], ubv[1], ubv[2], ubv[3]);
      *reinterpret_cast<float4*>(&s_ubar[r + 4]) =
          make_float4(ubv[4], ubv[5], ubv[6], ubv[7]);
#pragma unroll
      for (int k = 0; k < 8; ++k) ppr[k] = pcr[k];
    }

    __threadfence_block();
    __syncthreads();         // new ubar visible before next dual step
  }
  // d_out already holds the final u.
}

extern "C" void kernel_launch(void* const* d_in, const int* in_sizes, int n_in,
                              void* d_out, int out_size, void* d_ws, size_t ws_size,
                              hipStream_t stream) {
  (void)in_sizes; (void)n_in; (void)out_size; (void)ws_size;
  const float* f   = (const float*)d_in[0];
  const float* lam = (const float*)d_in[1];
  float*       out = (float*)d_out;
  float*       p   = (float*)d_ws;                         // 2MB
  float*       q   = (float*)d_ws + (size_t)8 * IMG_ELEMS; // 2MB (ws >= 4MB)

  // Allow >64KB dynamic LDS (idempotent, capture-safe).
  (void)hipFuncSetAttribute((const void*)tv_chambolle_pock,
                            hipFuncAttributeMaxDynamicSharedMemorySize,
                            (int)LDS_BYTES);

  hipLaunchKernelGGL(tv_chambolle_pock, dim3(8), dim3(1024), LDS_BYTES, stream,
                     f, lam, out, p, q);
}